// GIN_9388798509633
// MI455X (gfx1250) — compile-verified
//
#include <hip/hip_runtime.h>
#include <math.h>

typedef float v2f __attribute__((ext_vector_type(2)));
typedef float v8f __attribute__((ext_vector_type(8)));

static constexpr int kH   = 64;   // hidden size
static constexpr int kOUT = 16;   // output size
static constexpr int kL   = 5;    // layers
static constexpr float kGINEPS = 0.3f;
static constexpr float kBNEPS  = 1e-5f;

static __device__ __forceinline__ v8f wmma4(v2f a, v2f b, v8f c) {
  // D = A(16x4,f32) * B(4x16,f32) + C(16x16,f32), fp32-exact
  return __builtin_amdgcn_wmma_f32_16x16x4_f32(false, a, false, b, (short)0, c,
                                               false, false);
}

// ---------------------------------------------------------------------------
// agg = (1+eps) * h ; also zero BN stat accumulators (128 floats)
// ---------------------------------------------------------------------------
__global__ void gin_init_agg(const float* __restrict__ h, float* __restrict__ agg,
                             float* __restrict__ stats, long total) {
  long i = (long)blockIdx.x * blockDim.x + threadIdx.x;
  if (i < 2 * kH) stats[i] = 0.0f;
  if (i < total) agg[i] = (1.0f + kGINEPS) * h[i];
}

// ---------------------------------------------------------------------------
// agg[dst] += h[src]  (8 threads per edge, 8 columns each, float4 loads)
// ---------------------------------------------------------------------------
__global__ void gin_edge_scatter(const float* __restrict__ h,
                                 const int* __restrict__ ei,
                                 float* __restrict__ agg, int E) {
  long t = (long)blockIdx.x * blockDim.x + threadIdx.x;
  long total = (long)E * 8;
  if (t >= total) return;
  int e  = (int)(t >> 3);
  int c0 = ((int)t & 7) * 8;
  int s = ei[e];       // src row of edge_index
  int d = ei[E + e];   // dst row of edge_index
  const float4* hp = (const float4*)(h + (long)s * kH + c0);
  float4 v0 = hp[0];
  float4 v1 = hp[1];
  float* ap = agg + (long)d * kH + c0;
  atomicAdd(ap + 0, v0.x); atomicAdd(ap + 1, v0.y);
  atomicAdd(ap + 2, v0.z); atomicAdd(ap + 3, v0.w);
  atomicAdd(ap + 4, v1.x); atomicAdd(ap + 5, v1.y);
  atomicAdd(ap + 6, v1.z); atomicAdd(ap + 7, v1.w);
}

// ---------------------------------------------------------------------------
// Shared helper: stage W (row-major [k][n]) into LDS with K-pair interleave:
//   sW[(k>>1)*2*kH + n*2 + (k&1)]  ->  B fragment = single aligned b64 load
// ---------------------------------------------------------------------------
static __device__ __forceinline__ void stage_w_pairs(float* sW,
                                                     const float* __restrict__ W,
                                                     int tid) {
  for (int i = tid; i < kH * kH; i += 256) {
    int r = i >> 6;      // k
    int c = i & 63;      // n
    sW[(r >> 1) * (2 * kH) + c * 2 + (r & 1)] = W[i];
  }
}

// ---------------------------------------------------------------------------
// y = agg @ W + b  (fp32 WMMA), plus per-column sum / sum-of-squares for BN
// block = 256 threads (8 waves), 128 rows per block, full 64 columns
// ---------------------------------------------------------------------------
__global__ __launch_bounds__(256) void gin_gemm1(
    const float* __restrict__ X, const float* __restrict__ W,
    const float* __restrict__ bias, float* __restrict__ Y,
    float* __restrict__ stats, int N) {
  __shared__ float sW[kH * kH];   // K-pair interleaved
  __shared__ float sB[kH];
  __shared__ float sSum[kH];
  __shared__ float sSq[kH];
  int tid = threadIdx.x;
  stage_w_pairs(sW, W, tid);
  if (tid < kH) { sB[tid] = bias[tid]; sSum[tid] = 0.0f; sSq[tid] = 0.0f; }
  __syncthreads();

  int wave = tid >> 5;
  int lane = tid & 31;
  int lh   = lane >> 4;   // K-half select
  int l15  = lane & 15;
  int m0   = blockIdx.x * 128 + wave * 16;
  int arow = m0 + l15;
  int arow_c = arow < N ? arow : (N - 1);   // clamp: no divergence in hot loop
  const float* xrow = X + (long)arow_c * kH + 2 * lh;

  v8f acc0 = {}, acc1 = {}, acc2 = {}, acc3 = {};
#pragma unroll
  for (int k = 0; k < kH; k += 4) {
    v2f a = *(const v2f*)(xrow + k);               // cols kb, kb+1
    int kp = (k >> 1) + lh;                        // K-pair index
    const v2f* bp = (const v2f*)(sW + kp * (2 * kH));
    v2f b0 = bp[ 0 + l15];
    v2f b1 = bp[16 + l15];
    v2f b2 = bp[32 + l15];
    v2f b3 = bp[48 + l15];
    acc0 = wmma4(a, b0, acc0);
    acc1 = wmma4(a, b1, acc1);
    acc2 = wmma4(a, b2, acc2);
    acc3 = wmma4(a, b3, acc3);
  }

#pragma unroll
  for (int v = 0; v < 8; ++v) {
    int row = m0 + 8 * lh + v;
    if (row < N) {
      float vals[4] = {acc0[v], acc1[v], acc2[v], acc3[v]};
#pragma unroll
      for (int j = 0; j < 4; ++j) {
        int col = j * 16 + l15;
        float val = vals[j] + sB[col];
        Y[(long)row * kH + col] = val;
        atomicAdd(&sSum[col], val);
        atomicAdd(&sSq[col], val * val);
      }
    }
  }
  __syncthreads();
  if (tid < kH) {
    atomicAdd(&stats[tid], sSum[tid]);
    atomicAdd(&stats[kH + tid], sSq[tid]);
  }
}

// ---------------------------------------------------------------------------
// scale = gamma*rsqrt(var+eps), shift = beta - mu*scale   (one tiny block)
// ---------------------------------------------------------------------------
__global__ void gin_bn_prep(const float* __restrict__ stats,
                            const float* __restrict__ gamma,
                            const float* __restrict__ beta,
                            float* __restrict__ ss, float invN) {
  int c = threadIdx.x;
  if (c < kH) {
    float mu  = stats[c] * invN;
    float var = stats[kH + c] * invN - mu * mu;
    float sc  = gamma[c] * rsqrtf(var + kBNEPS);
    ss[c]      = sc;
    ss[kH + c] = beta[c] - mu * sc;
  }
}

// ---------------------------------------------------------------------------
// hn = tanh( tanh(scale*y + shift) @ W2 + b2 )  — BN+tanh fused into A-load
// ---------------------------------------------------------------------------
__global__ __launch_bounds__(256) void gin_gemm2(
    const float* __restrict__ Y, const float* __restrict__ ss,
    const float* __restrict__ W, const float* __restrict__ bias,
    float* __restrict__ Hn, int N) {
  __shared__ float sW[kH * kH];   // K-pair interleaved
  __shared__ float sB[kH];
  __shared__ float sScale[kH];
  __shared__ float sShift[kH];
  int tid = threadIdx.x;
  stage_w_pairs(sW, W, tid);
  if (tid < kH) {
    sB[tid]     = bias[tid];
    sScale[tid] = ss[tid];
    sShift[tid] = ss[kH + tid];
  }
  __syncthreads();

  int wave = tid >> 5;
  int lane = tid & 31;
  int lh   = lane >> 4;
  int l15  = lane & 15;
  int m0   = blockIdx.x * 128 + wave * 16;
  int arow = m0 + l15;
  int arow_c = arow < N ? arow : (N - 1);
  const float* yrow = Y + (long)arow_c * kH + 2 * lh;

  v8f acc0 = {}, acc1 = {}, acc2 = {}, acc3 = {};
#pragma unroll
  for (int k = 0; k < kH; k += 4) {
    int kb = k + 2 * lh;
    v2f p = *(const v2f*)(yrow + k);
    v2f a;
    a.x = tanhf(sScale[kb] * p.x + sShift[kb]);
    a.y = tanhf(sScale[kb + 1] * p.y + sShift[kb + 1]);
    int kp = (k >> 1) + lh;
    const v2f* bp = (const v2f*)(sW + kp * (2 * kH));
    v2f b0 = bp[ 0 + l15];
    v2f b1 = bp[16 + l15];
    v2f b2 = bp[32 + l15];
    v2f b3 = bp[48 + l15];
    acc0 = wmma4(a, b0, acc0);
    acc1 = wmma4(a, b1, acc1);
    acc2 = wmma4(a, b2, acc2);
    acc3 = wmma4(a, b3, acc3);
  }

#pragma unroll
  for (int v = 0; v < 8; ++v) {
    int row = m0 + 8 * lh + v;
    if (row < N) {
      float vals[4] = {acc0[v], acc1[v], acc2[v], acc3[v]};
#pragma unroll
      for (int j = 0; j < 4; ++j) {
        int col = j * 16 + l15;
        Hn[(long)row * kH + col] = tanhf(vals[j] + sB[col]);
      }
    }
  }
}

// ---------------------------------------------------------------------------
// pooled zero + pooled[batch[i]] += h[i]
// ---------------------------------------------------------------------------
__global__ void gin_zero(float* __restrict__ p, int n) {
  int i = blockIdx.x * blockDim.x + threadIdx.x;
  if (i < n) p[i] = 0.0f;
}

__global__ void gin_pool(const float* __restrict__ h, const int* __restrict__ batch,
                         float* __restrict__ pooled, int N) {
  long t = (long)blockIdx.x * blockDim.x + threadIdx.x;
  long total = (long)N * 8;
  if (t >= total) return;
  int node = (int)(t >> 3);
  int c0   = ((int)t & 7) * 8;
  int g = batch[node];
  const float4* hp = (const float4*)(h + (long)node * kH + c0);
  float4 v0 = hp[0];
  float4 v1 = hp[1];
  float* pp = pooled + (long)g * kH + c0;
  atomicAdd(pp + 0, v0.x); atomicAdd(pp + 1, v0.y);
  atomicAdd(pp + 2, v0.z); atomicAdd(pp + 3, v0.w);
  atomicAdd(pp + 4, v1.x); atomicAdd(pp + 5, v1.y);
  atomicAdd(pp + 6, v1.z); atomicAdd(pp + 7, v1.w);
}

// ---------------------------------------------------------------------------
// out = sigmoid(pooled @ Wout + bout)   [G,64]x[64,16] — WMMA head
// ---------------------------------------------------------------------------
__global__ __launch_bounds__(256) void gin_final(
    const float* __restrict__ P, const float* __restrict__ Wout,
    const float* __restrict__ bout, float* __restrict__ out, int G) {
  __shared__ float sW[kH * kOUT];   // K-pair interleaved: [(k>>1)][n*2 + (k&1)]
  __shared__ float sB[kOUT];
  int tid = threadIdx.x;
  for (int i = tid; i < kH * kOUT; i += 256) {
    int r = i >> 4;       // k
    int c = i & 15;       // n
    sW[(r >> 1) * (2 * kOUT) + c * 2 + (r & 1)] = Wout[i];
  }
  if (tid < kOUT) sB[tid] = bout[tid];
  __syncthreads();

  int wave = tid >> 5;
  int lane = tid & 31;
  int lh   = lane >> 4;
  int l15  = lane & 15;
  int m0   = blockIdx.x * 128 + wave * 16;
  int arow = m0 + l15;
  int arow_c = arow < G ? arow : (G - 1);
  const float* prow = P + (long)arow_c * kH + 2 * lh;

  v8f acc = {};
#pragma unroll
  for (int k = 0; k < kH; k += 4) {
    v2f a = *(const v2f*)(prow + k);
    int kp = (k >> 1) + lh;
    v2f b = *(const v2f*)(sW + kp * (2 * kOUT) + l15 * 2);
    acc = wmma4(a, b, acc);
  }
#pragma unroll
  for (int v = 0; v < 8; ++v) {
    int row = m0 + 8 * lh + v;
    if (row < G) {
      float val = acc[v] + sB[l15];
      out[(long)row * kOUT + l15] = 1.0f / (1.0f + expf(-val));
    }
  }
}

// ---------------------------------------------------------------------------
extern "C" void kernel_launch(void* const* d_in, const int* in_sizes, int n_in,
                              void* d_out, int out_size, void* d_ws, size_t ws_size,
                              hipStream_t stream) {
  const float* x     = (const float*)d_in[0];
  const int*   ei    = (const int*)d_in[1];
  const int*   batch = (const int*)d_in[2];
  const float* W1    = (const float*)d_in[3];
  const float* b1    = (const float*)d_in[4];
  const float* gamma = (const float*)d_in[5];
  const float* beta  = (const float*)d_in[6];
  const float* W2    = (const float*)d_in[7];
  const float* b2    = (const float*)d_in[8];
  const float* Wout  = (const float*)d_in[9];
  const float* bout  = (const float*)d_in[10];
  float* out = (float*)d_out;

  const int N = in_sizes[0] / kH;
  const int E = in_sizes[1] / 2;
  const int G = out_size / kOUT;

  // workspace layout (fp32): two N*H ping-pong buffers, BN stats, scale/shift,
  // pooled accumulator
  float* ws0    = (float*)d_ws;
  float* ws1    = ws0 + (size_t)N * kH;
  float* stats  = ws1 + (size_t)N * kH;   // 2*kH: colsum, colsumsq
  float* ss     = stats + 2 * kH;         // 2*kH: scale, shift
  float* pooled = ss + 2 * kH;            // G*kH

  const long totNH = (long)N * kH;
  const int  nhBlocks = (int)((totNH + 255) / 256);
  const long totE8 = (long)E * 8;
  const int  eBlocks = (int)((totE8 + 255) / 256);
  const int  gemmBlocks = (N + 127) / 128;
  const float invN = 1.0f / (float)N;

  const float* h = x;
  for (int l = 0; l < kL; ++l) {
    float* agg = (h == ws0) ? ws1 : ws0;
    float* y   = (h == x) ? ((agg == ws0) ? ws1 : ws0) : (float*)h;
    float* hn  = agg;  // agg's buffer is dead by the time gemm2 writes

    gin_init_agg<<<nhBlocks, 256, 0, stream>>>(h, agg, stats, totNH);
    gin_edge_scatter<<<eBlocks, 256, 0, stream>>>(h, ei, agg, E);
    gin_gemm1<<<gemmBlocks, 256, 0, stream>>>(agg, W1 + (size_t)l * kH * kH,
                                              b1 + (size_t)l * kH, y, stats, N);
    gin_bn_prep<<<1, 64, 0, stream>>>(stats, gamma + (size_t)l * kH,
                                      beta + (size_t)l * kH, ss, invN);
    gin_gemm2<<<gemmBlocks, 256, 0, stream>>>(y, ss, W2 + (size_t)l * kH * kH,
                                              b2 + (size_t)l * kH, hn, N);
    h = hn;
  }

  const int pn = G * kH;
  gin_zero<<<(pn + 255) / 256, 256, 0, stream>>>(pooled, pn);
  const long totP = (long)N * 8;
  gin_pool<<<(int)((totP + 255) / 256), 256, 0, stream>>>(h, batch, pooled, N);
  gin_final<<<(G + 127) / 128, 256, 0, stream>>>(pooled, Wout, bout, out, G);
}